// LSTMPointerNet_22007412425124
// MI455X (gfx1250) — compile-verified
//
#include <hip/hip_runtime.h>
#include <hip/hip_bf16.h>

// ---------------------------------------------------------------------------
// LSTMPointerNet for MI455X (gfx1250, wave32, WMMA).
// All GEMMs use V_WMMA_F32_16X16X4_F32 (fp32 A/B, fp32 acc) -> exact fp32.
// GEMM operands are padded to 16-row multiples so there is NO per-lane
// predication anywhere in a WMMA loop (EXEC all-ones, no saveexec traffic).
// ---------------------------------------------------------------------------

typedef __attribute__((ext_vector_type(2))) float v2f;
typedef __attribute__((ext_vector_type(8))) float v8f;

static constexpr int kB = 8;      // batch
static constexpr int kN = 512;    // memory slots
static constexpr int kT = 64;     // lstm_in time steps
static constexpr int kS = 65;     // T+1
static constexpr int kD = 256;    // input dim
static constexpr int kH = 256;    // hidden dim
static constexpr int kG = 1024;   // 4*H gates
static constexpr int kMq = 528;   // kB*kS=520 padded to multiple of 16

#define WMMA_F32X4(a, b, c) \
  __builtin_amdgcn_wmma_f32_16x16x4_f32(false, (a), false, (b), (short)0, (c), false, false)

// ---------------------------------------------------------------------------
// fp32 WMMA GEMM:  C[M,256] = A[M,K] * B[K,256], both row-major, M % 16 == 0.
// One 16x64 tile per wave (4 accumulators): each A v2f load feeds 4 WMMAs.
// ---------------------------------------------------------------------------
__global__ __launch_bounds__(256) void wmma_gemm_f32(
    const float* __restrict__ A, const float* __restrict__ Bmat,
    float* __restrict__ C, int Mt /* = M/16 */, int K) {
  const int lane = threadIdx.x & 31;
  const int gw = blockIdx.x * 8 + (threadIdx.x >> 5);
  if (gw >= Mt * 4) return;           // wave-uniform exit
  const int m0 = (gw >> 2) << 4;
  const int n0 = (gw & 3) << 6;       // 64-wide column group
  const int half = lane >> 4;         // K sub-pair selector (16x4 A layout)
  const int l15 = lane & 15;

  const float* ap = A + (m0 + l15) * K + 2 * half;         // A[row][ka..ka+1]
  const float* bp = Bmat + 2 * half * kH + n0 + l15;       // B[ka][col]

  v8f a0 = {0.f, 0.f, 0.f, 0.f, 0.f, 0.f, 0.f, 0.f};
  v8f a1 = a0, a2 = a0, a3 = a0;
#pragma unroll 4
  for (int k = 0; k < K; k += 4) {
    const v2f av = *(const v2f*)ap;                        // global_load_b64
    v2f b0, b1, b2, b3;
    b0[0] = bp[0];       b0[1] = bp[kH];                   // imm-offset loads
    b1[0] = bp[16];      b1[1] = bp[kH + 16];
    b2[0] = bp[32];      b2[1] = bp[kH + 32];
    b3[0] = bp[48];      b3[1] = bp[kH + 48];
    a0 = WMMA_F32X4(av, b0, a0);
    a1 = WMMA_F32X4(av, b1, a1);
    a2 = WMMA_F32X4(av, b2, a2);
    a3 = WMMA_F32X4(av, b3, a3);
    ap += 4;
    bp += 4 * kH;
  }
#pragma unroll
  for (int v = 0; v < 8; ++v) {
    float* cr = C + (m0 + v + 8 * half) * kH + n0 + l15;   // M = v + 8*half
    cr[0] = a0[v];
    cr[16] = a1[v];
    cr[32] = a2[v];
    cr[48] = a3[v];
  }
}

// ---------------------------------------------------------------------------
// Persistent 2-layer LSTM, batch 8 (padded to M=16), hidden 256.
// One workgroup: 1024 threads = 32 waves; wave w owns gate-column tiles
// {2w, 2w+1} of the 64 16-wide tiles covering 4H=1024 gate outputs.
// A rows 8..15 are read clamped (l15&7): they only affect C rows 8..15,
// which are discarded, so no masking is needed -> plain ds_load_b64.
// ---------------------------------------------------------------------------
__global__ __launch_bounds__(1024) void lstm_wmma_kernel(
    const float* __restrict__ lstm_in,  // [B,T,D]
    const float* __restrict__ init_h,   // [L,H]
    const float* __restrict__ init_c,   // [L,H]
    const float* __restrict__ init_i,   // [D]
    const float* __restrict__ w_ih,     // [L,4H,D]
    const float* __restrict__ w_hh,     // [L,4H,H]
    const float* __restrict__ b_ih,     // [L,4H]
    const float* __restrict__ b_hh,     // [L,4H]
    float* __restrict__ query) {        // [B,S,H]
  __shared__ float xbuf[kB * kD];       // layer-0 input          (8 KB)
  __shared__ float hbuf[2][kB * kH];    // hidden state per layer (16 KB)
  __shared__ float gbuf[kB * kG];       // gate pre-activations   (32 KB)

  const int tid = threadIdx.x;
  const int lane = tid & 31;
  const int wave = tid >> 5;
  const int half = lane >> 4;
  const int l15 = lane & 15;
  const int arow = (l15 & 7) * kD;      // clamped A row base (kD == kH)

  // Each thread owns 2 (row, hidden) activation slots of the 8x256 state.
  const int e0 = tid, e1 = tid + 1024;
  const int r0 = e0 >> 8, j0 = e0 & 255;
  const int r1 = e1 >> 8, j1 = e1 & 255;
  float creg[2][2];
  for (int l = 0; l < 2; ++l) {
    hbuf[l][e0] = init_h[l * kH + j0];
    hbuf[l][e1] = init_h[l * kH + j1];
    creg[l][0] = init_c[l * kH + j0];
    creg[l][1] = init_c[l * kH + j1];
  }
  __syncthreads();

  for (int t = 0; t < kS; ++t) {
    // stage x_t (t==0: init_i broadcast; else lstm_in[:, t-1, :])
    if (t == 0) {
      xbuf[e0] = init_i[j0];
      xbuf[e1] = init_i[j1];
    } else {
      xbuf[e0] = lstm_in[r0 * kT * kD + (t - 1) * kD + j0];
      xbuf[e1] = lstm_in[r1 * kT * kD + (t - 1) * kD + j1];
    }
    __syncthreads();

    for (int l = 0; l < 2; ++l) {
      const float* Ain = (l == 0) ? xbuf : hbuf[0];
      const float* Ah = hbuf[l];

      for (int tt = 0; tt < 2; ++tt) {
        const int col = (((wave << 1) + tt) << 4) + l15;   // gate column
        v8f acc = {0.f, 0.f, 0.f, 0.f, 0.f, 0.f, 0.f, 0.f};
        {  // g += x @ Wih^T : B[k][col] = Wih[col][k] (adjacent pair -> b64)
          const float* ap = Ain + arow + 2 * half;
          const float* wp = w_ih + (l * kG + col) * kD + 2 * half;
#pragma unroll 4
          for (int k = 0; k < kD; k += 4) {
            const v2f av = *(const v2f*)ap;                // ds_load_b64
            const v2f bv = *(const v2f*)wp;                // global_load_b64
            acc = WMMA_F32X4(av, bv, acc);
            ap += 4;
            wp += 4;
          }
        }
        {  // g += h @ Whh^T
          const float* ap = Ah + arow + 2 * half;
          const float* wp = w_hh + (l * kG + col) * kH + 2 * half;
#pragma unroll 4
          for (int k = 0; k < kH; k += 4) {
            const v2f av = *(const v2f*)ap;
            const v2f bv = *(const v2f*)wp;
            acc = WMMA_F32X4(av, bv, acc);
            ap += 4;
            wp += 4;
          }
        }
        if (half == 0) {  // rows 0..7 live in the half==0 lanes
#pragma unroll
          for (int v = 0; v < 8; ++v) gbuf[v * kG + col] = acc[v];
        }
      }
      __syncthreads();

      // activations: i,f,g,o at columns j, j+256, j+512, j+768
#pragma unroll
      for (int s = 0; s < 2; ++s) {
        const int r = s ? r1 : r0, j = s ? j1 : j0;
        const int gb = r * kG + j, bb = l * kG + j;
        const float gi = gbuf[gb] + b_ih[bb] + b_hh[bb];
        const float gf = gbuf[gb + 256] + b_ih[bb + 256] + b_hh[bb + 256];
        const float gg = gbuf[gb + 512] + b_ih[bb + 512] + b_hh[bb + 512];
        const float go = gbuf[gb + 768] + b_ih[bb + 768] + b_hh[bb + 768];
        const float ig = 1.0f / (1.0f + __expf(-gi));
        const float fg = 1.0f / (1.0f + __expf(-gf));
        const float og = 1.0f / (1.0f + __expf(-go));
        const float cn = fg * creg[l][s] + ig * tanhf(gg);
        creg[l][s] = cn;
        const float hn = og * tanhf(cn);
        hbuf[l][r * kH + j] = hn;
        if (l == 1) query[r * kS * kH + t * kH + j] = hn;
      }
      __syncthreads();
    }
  }
}

// ---------------------------------------------------------------------------
// Fused hop attention: score -> mask -> softmax(N) -> context.
// One block per (b, s); 256 threads. ctx overwrites the query buffer.
// ---------------------------------------------------------------------------
__global__ __launch_bounds__(256) void hop_attend_kernel(
    const float* __restrict__ feat,  // hop_feat [B,N,H]
    const float* __restrict__ q,     // query @ hop_wq [B(S),H] padded rows
    const float* __restrict__ vvec,  // [H]
    const int* __restrict__ mem_sizes,
    float* __restrict__ ctx_out) {   // [B,S,H]
  const int b = blockIdx.x / kS;
  const int s = blockIdx.x % kS;
  const int tid = threadIdx.x;
  __shared__ float qs[kH], vs[kH], prob[kN], red[256];

  qs[tid] = q[(b * kS + s) * kH + tid];
  vs[tid] = vvec[tid];
  __syncthreads();

  const int msz = mem_sizes[b];
  float scl[2];
#pragma unroll
  for (int i = 0; i < 2; ++i) {
    const int n = tid + i * 256;
    const float4* f4 = (const float4*)(feat + b * kN * kH + n * kH);
    float acc = 0.0f;
#pragma unroll 4
    for (int h = 0; h < kH / 4; ++h) {
      const float4 fv = f4[h];                             // global_load_b128
      acc += vs[4 * h + 0] * tanhf(fv.x + qs[4 * h + 0]);
      acc += vs[4 * h + 1] * tanhf(fv.y + qs[4 * h + 1]);
      acc += vs[4 * h + 2] * tanhf(fv.z + qs[4 * h + 2]);
      acc += vs[4 * h + 3] * tanhf(fv.w + qs[4 * h + 3]);
    }
    scl[i] = (n < msz) ? acc : -1e18f;
  }
  // block max
  red[tid] = fmaxf(scl[0], scl[1]);
  __syncthreads();
  for (int st = 128; st > 0; st >>= 1) {
    if (tid < st) red[tid] = fmaxf(red[tid], red[tid + st]);
    __syncthreads();
  }
  const float mx = red[0];
  __syncthreads();
  // block sum of exp
  const float ex0 = __expf(scl[0] - mx), ex1 = __expf(scl[1] - mx);
  red[tid] = ex0 + ex1;
  __syncthreads();
  for (int st = 128; st > 0; st >>= 1) {
    if (tid < st) red[tid] += red[tid + st];
    __syncthreads();
  }
  const float inv = 1.0f / red[0];
  __syncthreads();
  prob[tid] = ex0 * inv;
  prob[tid + 256] = ex1 * inv;
  __syncthreads();
  // context: thread tid == hidden index h
  float ctx = 0.0f;
  const float* fb = feat + b * kN * kH;
#pragma unroll 4
  for (int n = 0; n < kN; ++n) ctx += prob[n] * fb[n * kH + tid];
  ctx_out[(b * kS + s) * kH + tid] = ctx;
}

// ---------------------------------------------------------------------------
// Final additive score (no mask / softmax):
// out[b,s,n] = sum_h attn_v[h] * tanh(attn_feat[b,n,h] + q2[b,s,h])
// ---------------------------------------------------------------------------
__global__ __launch_bounds__(256) void final_score_kernel(
    const float* __restrict__ feat,  // attn_feat [B,N,H]
    const float* __restrict__ q2,    // ctx @ attn_wq, padded rows
    const float* __restrict__ vvec,  // [H]
    float* __restrict__ out) {       // [B,S,N]
  const int b = blockIdx.x / kS;
  const int s = blockIdx.x % kS;
  const int tid = threadIdx.x;
  __shared__ float qs[kH], vs[kH];
  qs[tid] = q2[(b * kS + s) * kH + tid];
  vs[tid] = vvec[tid];
  __syncthreads();
#pragma unroll
  for (int i = 0; i < 2; ++i) {
    const int n = tid + i * 256;
    const float4* f4 = (const float4*)(feat + b * kN * kH + n * kH);
    float acc = 0.0f;
#pragma unroll 4
    for (int h = 0; h < kH / 4; ++h) {
      const float4 fv = f4[h];
      acc += vs[4 * h + 0] * tanhf(fv.x + qs[4 * h + 0]);
      acc += vs[4 * h + 1] * tanhf(fv.y + qs[4 * h + 1]);
      acc += vs[4 * h + 2] * tanhf(fv.z + qs[4 * h + 2]);
      acc += vs[4 * h + 3] * tanhf(fv.w + qs[4 * h + 3]);
    }
    out[(b * kS + s) * kN + n] = acc;
  }
}

// Zero the padding rows (520..527) of the query buffer each launch.
__global__ __launch_bounds__(256) void zero_pad_kernel(float* __restrict__ p) {
  p[blockIdx.x * 256 + threadIdx.x] = 0.0f;
}

// ---------------------------------------------------------------------------
extern "C" void kernel_launch(void* const* d_in, const int* in_sizes, int n_in,
                              void* d_out, int out_size, void* d_ws,
                              size_t ws_size, hipStream_t stream) {
  const float* attn_mem = (const float*)d_in[0];
  const int* mem_sizes = (const int*)d_in[1];
  const float* lstm_in = (const float*)d_in[2];
  const float* init_h = (const float*)d_in[3];
  const float* init_c = (const float*)d_in[4];
  const float* init_i = (const float*)d_in[5];
  const float* w_ih = (const float*)d_in[6];
  const float* w_hh = (const float*)d_in[7];
  const float* b_ih = (const float*)d_in[8];
  const float* b_hh = (const float*)d_in[9];
  const float* attn_wm = (const float*)d_in[10];
  const float* attn_wq = (const float*)d_in[11];
  const float* attn_v = (const float*)d_in[12];
  const float* hop_wm = (const float*)d_in[13];
  const float* hop_wq = (const float*)d_in[14];
  const float* hop_v = (const float*)d_in[15];
  float* out = (float*)d_out;

  // workspace: attn_feat | hop_feat | query (later ctx) | qbuf  (rows padded)
  float* ws = (float*)d_ws;
  float* attn_feat = ws;                      // B*N*H
  float* hop_feat = attn_feat + kB * kN * kH; // B*N*H
  float* query = hop_feat + kB * kN * kH;     // kMq*H (520 valid + 8 pad)
  float* qbuf = query + kMq * kH;             // kMq*H

  auto gemm = [&](const float* A, const float* Bm, float* C, int Mt, int K) {
    const int blocks = (Mt * 4 + 7) / 8;  // 8 waves / block, 4 col-groups
    hipLaunchKernelGGL(wmma_gemm_f32, dim3(blocks), dim3(256), 0, stream, A,
                       Bm, C, Mt, K);
  };

  // 0) zero the 8 padding rows of `query` (8*256 = 2048 floats)
  hipLaunchKernelGGL(zero_pad_kernel, dim3(8), dim3(256), 0, stream,
                     query + kB * kS * kH);

  // 1) feature projections
  gemm(attn_mem, attn_wm, attn_feat, (kB * kN) / 16, kD);
  gemm(attn_mem, hop_wm, hop_feat, (kB * kN) / 16, kD);

  // 2) persistent 2-layer LSTM -> query [B,S,H]
  hipLaunchKernelGGL(lstm_wmma_kernel, dim3(1), dim3(1024), 0, stream, lstm_in,
                     init_h, init_c, init_i, w_ih, w_hh, b_ih, b_hh, query);

  // 3) hop: qbuf = query @ hop_wq ; attend ; ctx overwrites `query`
  gemm(query, hop_wq, qbuf, kMq / 16, kH);
  hipLaunchKernelGGL(hop_attend_kernel, dim3(kB * kS), dim3(256), 0, stream,
                     hop_feat, qbuf, hop_v, mem_sizes, query);

  // 4) final: qbuf = ctx @ attn_wq ; raw additive score -> d_out
  gemm(query, attn_wq, qbuf, kMq / 16, kH);
  hipLaunchKernelGGL(final_score_kernel, dim3(kB * kS), dim3(256), 0, stream,
                     attn_feat, qbuf, attn_v, out);
}